// ExternalInteraction_9079560863791
// MI455X (gfx1250) — compile-verified
//
#include <hip/hip_runtime.h>

// ---------------------------------------------------------------------------
// ExternalInteraction fused kernel for MI455X (gfx1250, wave32).
//
// out_user[b,u] = user[b,u] * sum_i image[b,i]
// out_img [b,i] = image[b,i] * sum_u user[b,u]
//
// Bandwidth-bound: 128 MiB total traffic -> ~5.8us at 23.3 TB/s.
// One 256-thread block per batch row; data stays in VGPRs across the
// reduction; the 256-partial reduction is done with chained
// V_WMMA_F32_16X16X4_F32 against a ones matrix (row-sum == A x 1).
// ---------------------------------------------------------------------------

typedef __attribute__((ext_vector_type(2))) float v2f;
typedef __attribute__((ext_vector_type(4))) float v4f;
typedef __attribute__((ext_vector_type(8))) float v8f;

#define ROWLEN 4096
#define TPB    256

// Reduce 256 floats living in LDS to a single scalar using
// V_WMMA_F32_16X16X4_F32 with B = ones (so D[m,n] = rowsum_m, any n).
// A-matrix 16x4 fp32 lane layout (ISA 7.12.2): lanes 0-15 hold A[m, 0..1],
// lanes 16-31 hold A[m, 2..3], m = lane % 16, 2 VGPRs per lane.
// We map A_i[m,k] = lds[i*64 + m*4 + k], i = 0..3 accumulated into C,
// which covers all 256 entries exactly once.
// Must be called with all 32 lanes of the wave active (EXEC all ones).
__device__ __forceinline__ float wmma_reduce256(const float* lds) {
  const int lane = threadIdx.x & 31;
  const int m    = lane & 15;
  const int koff = (lane < 16) ? 0 : 2;

  v2f bones;
  bones.x = 1.0f;
  bones.y = 1.0f;

  v8f acc = {};
#pragma unroll
  for (int i = 0; i < 4; ++i) {
    // 8-byte aligned: m*4 + koff is even.
    v2f a = *(const v2f*)(lds + i * 64 + m * 4 + koff);
    acc = __builtin_amdgcn_wmma_f32_16x16x4_f32(
        /*neg_a=*/false, a, /*neg_b=*/false, bones,
        /*c_mod=*/(short)0, acc, /*reuse_a=*/false, /*reuse_b=*/false);
  }

  // D layout: VGPR r, lanes 0-15 -> D[r, lane] (M=r), lanes 16-31 -> D[r+8, .].
  // Columns are identical (B = ones), so:
  //   lanes 0-15  hold rowsums for M=0..7  across acc[0..7]
  //   lanes 16-31 hold rowsums for M=8..15 across acc[0..7]
  float s = acc[0] + acc[1] + acc[2] + acc[3] +
            acc[4] + acc[5] + acc[6] + acc[7];
  float lo = __shfl(s, 0, 32);   // sum of rowsums M=0..7
  float hi = __shfl(s, 16, 32);  // sum of rowsums M=8..15
  return lo + hi;
}

__global__ __launch_bounds__(TPB) void ExternalInteraction_9079560863791_kernel(
    const float* __restrict__ user, const float* __restrict__ img,
    float* __restrict__ out_user, float* __restrict__ out_img) {
  __shared__ float su[TPB];
  __shared__ float si[TPB];

  const int  tid     = threadIdx.x;
  const long rowbase = (long)blockIdx.x * ROWLEN;
  const float* urow  = user + rowbase;
  const float* irow  = img + rowbase;

  // Each thread: 4 x float4 from each matrix, kept in registers.
  v4f uv[4], iv[4];
  float pu = 0.0f, pi = 0.0f;
#pragma unroll
  for (int j = 0; j < 4; ++j) {
    const int idx = (j * TPB + tid) * 4;  // coalesced b128 across the wave
    uv[j] = __builtin_nontemporal_load((const v4f*)(urow + idx));
    iv[j] = __builtin_nontemporal_load((const v4f*)(irow + idx));
    pu += uv[j].x + uv[j].y + uv[j].z + uv[j].w;
    pi += iv[j].x + iv[j].y + iv[j].z + iv[j].w;
  }

  su[tid] = pu;
  si[tid] = pi;
  __syncthreads();

  // Every wave reduces redundantly (no divergence -> EXEC all ones at WMMA;
  // no second barrier / broadcast needed).
  const float usum = wmma_reduce256(su);
  const float isum = wmma_reduce256(si);

#pragma unroll
  for (int j = 0; j < 4; ++j) {
    const int idx = (j * TPB + tid) * 4;
    v4f ou = uv[j] * isum;  // user scaled by image row-sum
    v4f oi = iv[j] * usum;  // image scaled by user row-sum
    __builtin_nontemporal_store(ou, (v4f*)(out_user + rowbase + idx));
    __builtin_nontemporal_store(oi, (v4f*)(out_img + rowbase + idx));
  }
}

extern "C" void kernel_launch(void* const* d_in, const int* in_sizes, int n_in,
                              void* d_out, int out_size, void* d_ws, size_t ws_size,
                              hipStream_t stream) {
  (void)in_sizes; (void)n_in; (void)out_size; (void)d_ws; (void)ws_size;

  const float* user = (const float*)d_in[0];  // (2048, 4096) fp32
  const float* img  = (const float*)d_in[1];  // (2048, 4096) fp32

  const long B = 2048, U = 4096;
  float* out_user = (float*)d_out;            // first tuple element
  float* out_img  = (float*)d_out + B * U;    // second tuple element

  ExternalInteraction_9079560863791_kernel<<<dim3(B), dim3(TPB), 0, stream>>>(
      user, img, out_user, out_img);
}